// GraphVAE_17669495455921
// MI455X (gfx1250) — compile-verified
//
#include <hip/hip_runtime.h>
#include <hip/hip_bf16.h>

#define Nn 96
#define INF_ 64
#define HID 256
#define ZD 64
#define TRI 4656
#define MPM_ITERS 50
#define NE 2048

typedef float v2f __attribute__((ext_vector_type(2)));
typedef float v8f __attribute__((ext_vector_type(8)));

// ---------------------------------------------------------------------------
// Generic fp32 WMMA GEMM: C[M x Ncols] = A[M x K] * B[K x Ncols], row-major.
// One wave32 per 16x16 output tile; K stepped by 4 via V_WMMA_F32_16X16X4_F32.
// M, Ncols multiples of 16; K multiple of 4. EXEC all ones (32 threads/block).
// ---------------------------------------------------------------------------
__global__ __launch_bounds__(32) void k_wmma_gemm_f32(
    const float* __restrict__ A, const float* __restrict__ B,
    float* __restrict__ C, int Ncols, int K, int lda, int ldb, int ldc) {
  const int ntn  = Ncols >> 4;
  const int tm   = blockIdx.x / ntn;
  const int tn   = blockIdx.x % ntn;
  const int lane = threadIdx.x;      // 0..31
  const int half = lane >> 4;        // 0 or 1
  const int l16  = lane & 15;

  const float* Arow = A + (tm * 16 + l16) * lda;  // A frag: m = lane&15
  const float* Bcol = B + (tn * 16 + l16);        // B frag: n = lane&15
  v8f acc = {};
  for (int k0 = 0; k0 < K; k0 += 4) {
    const int ko = k0 + half * 2;   // lanes 16-31 carry K = +2,+3
    v2f a, b;
    a.x = Arow[ko];
    a.y = Arow[ko + 1];
    b.x = Bcol[ko * ldb];
    b.y = Bcol[(ko + 1) * ldb];
    acc = __builtin_amdgcn_wmma_f32_16x16x4_f32(
        /*neg_a=*/false, a, /*neg_b=*/false, b,
        /*c_mod=*/(short)0, acc, /*reuse_a=*/false, /*reuse_b=*/false);
  }
  float* Cout = C + (tn * 16 + l16);
#pragma unroll
  for (int r = 0; r < 8; ++r)
    Cout[(tm * 16 + r + half * 8) * ldc] = acc[r];
}

// ---------------------------------------------------------------------------
// Graph degree (deg[v] = #(dst==v) + 1 self-loop) and dis = rsqrt(deg)
// ---------------------------------------------------------------------------
__global__ void k_deg(const int* __restrict__ ei, float* __restrict__ deg,
                      float* __restrict__ dis) {
  int v = blockIdx.x * blockDim.x + threadIdx.x;
  if (v >= Nn) return;
  const int* dst = ei + NE;
  int c = 1;
  for (int e = 0; e < NE; ++e) c += (dst[e] == v) ? 1 : 0;
  float d = (float)c;
  deg[v] = d;
  dis[v] = rsqrtf(d);
}

// Dense sym-normalized adjacency An[d][s]; self-loops on diag, edges atomically
__global__ void k_An_init(const float* __restrict__ dis, float* __restrict__ An) {
  int t = blockIdx.x * blockDim.x + threadIdx.x;
  if (t >= Nn * Nn) return;
  int i = t / Nn, j = t % Nn;
  An[t] = (i == j) ? dis[i] * dis[i] : 0.0f;
}
__global__ void k_An_edges(const int* __restrict__ ei, const float* __restrict__ dis,
                           float* __restrict__ An) {
  int e = blockIdx.x * blockDim.x + threadIdx.x;
  if (e >= NE) return;
  int s = ei[e], d = ei[NE + e];
  atomicAdd(&An[d * Nn + s], dis[s] * dis[d]);
}

// bias + batchnorm (population var, eps=1e-5) + relu, per column
__global__ void k_bias_bn_relu(const float* __restrict__ Hin, const float* __restrict__ bias,
                               const float* __restrict__ gam, const float* __restrict__ bet,
                               float* __restrict__ Hout) {
  int c = blockIdx.x * blockDim.x + threadIdx.x;
  if (c >= HID) return;
  float bc = bias[c];
  float s = 0.0f, ss = 0.0f;
  for (int r = 0; r < Nn; ++r) {
    float v = Hin[r * HID + c] + bc;
    s += v; ss += v * v;
  }
  float m   = s / (float)Nn;
  float var = ss / (float)Nn - m * m;
  float inv = rsqrtf(var + 1e-5f);
  float g = gam[c], b = bet[c];
  for (int r = 0; r < Nn; ++r) {
    float v = Hin[r * HID + c] + bc;
    float o = (v - m) * inv * g + b;
    Hout[r * HID + c] = fmaxf(o, 0.0f);
  }
}

__global__ void k_pool(const float* __restrict__ H, float* __restrict__ g) {
  int c = blockIdx.x * blockDim.x + threadIdx.x;
  if (c >= HID) return;
  float s = 0.0f;
  for (int r = 0; r < Nn; ++r) s += H[r * HID + c];
  g[c] = s;
}

// mu / lv heads + reparameterization
__global__ void k_head(const float* __restrict__ g, const float* __restrict__ muw,
                       const float* __restrict__ mub, const float* __restrict__ lvw,
                       const float* __restrict__ lvb, const float* __restrict__ eps,
                       float* __restrict__ mu, float* __restrict__ lv,
                       float* __restrict__ z) {
  int t = blockIdx.x * blockDim.x + threadIdx.x;
  if (t >= ZD) return;
  float m = mub[t], l = lvb[t];
  for (int h = 0; h < HID; ++h) {
    float gh = g[h];
    m += gh * muw[h * ZD + t];
    l += gh * lvw[h * ZD + t];
  }
  mu[t] = m;
  lv[t] = l;
  z[t]  = m + eps[t] * expf(0.5f * l);
}

// decoder layer 1 -> zero-padded 16 x HID buffer (row 0 is the real row)
__global__ void k_dec1(const float* __restrict__ z, const float* __restrict__ w,
                       const float* __restrict__ b, float* __restrict__ hpad) {
  int t = blockIdx.x * blockDim.x + threadIdx.x;
  if (t >= 16 * HID) return;
  int r = t / HID, c = t % HID;
  float o = 0.0f;
  if (r == 0) {
    o = b[c];
    for (int k = 0; k < ZD; ++k) o += z[k] * w[k * HID + c];
    o = fmaxf(o, 0.0f);
  }
  hpad[t] = o;
}

__device__ __forceinline__ int triidx(int i, int j) {  // i <= j
  return i * Nn - (i * (i - 1)) / 2 + (j - i);
}

// symmetrized logits -> vec (upper tri, with bias) and B = sigmoid(logits)
__global__ void k_buildB(const float* __restrict__ vecP, const float* __restrict__ d2b,
                         float* __restrict__ vec, float* __restrict__ Bm) {
  int t = blockIdx.x * blockDim.x + threadIdx.x;
  if (t >= Nn * Nn) return;
  int i = t / Nn, j = t % Nn;
  int a = (i < j) ? i : j;
  int b = (i < j) ? j : i;
  int idx = triidx(a, b);
  float l = vecP[idx] + d2b[idx];   // row 0 of padded GEMM output
  if (i <= j) vec[idx] = l;
  Bm[t] = 1.0f / (1.0f + expf(-l));
}

__global__ void k_degAB(const float* __restrict__ Agt, const float* __restrict__ Bm,
                        float* __restrict__ degA, float* __restrict__ degB) {
  int v = blockIdx.x * blockDim.x + threadIdx.x;
  if (v >= Nn) return;
  float sa = 0.0f, sb = 0.0f;
  for (int j = 0; j < Nn; ++j) {
    sa += Agt[v * Nn + j];
    sb += Bm[v * Nn + j];
  }
  degA[v] = sa;
  degB[v] = sb;
}

// P, Q (off-diagonal rank-1 factors of S), D (diagonal block), X init
__global__ void k_pqdx(const float* __restrict__ Agt, const float* __restrict__ Bm,
                       const float* __restrict__ degA, const float* __restrict__ degB,
                       float* __restrict__ P, float* __restrict__ Q,
                       float* __restrict__ Dm, float* __restrict__ X) {
  int t = blockIdx.x * blockDim.x + threadIdx.x;
  if (t >= Nn * Nn) return;
  int i = t / Nn, j = t % Nn;
  float Adi = Agt[i * Nn + i], Adj = Agt[j * Nn + j];
  float Bdi = Bm[i * Nn + i],  Bdj = Bm[j * Nn + j];
  float off = (i == j) ? 0.0f : 1.0f;
  P[t]  = Agt[t] * Adi * Adj * off;
  Q[t]  = Bm[t] * Bdi * Bdj * off;
  Dm[t] = Adi * Bdj / (fabsf(degA[i] - degB[j]) + 1.0f);
  X[t]  = 1.0f / (float)Nn;
}

// M[j][a] = max_b Q[a][b]*X[j][b]   (rank-1 factorization of the N^4 max)
__global__ void k_maxM(const float* __restrict__ Q, const float* __restrict__ X,
                       float* __restrict__ Mb, float* __restrict__ nrm) {
  int t = blockIdx.x * blockDim.x + threadIdx.x;
  if (t == 0) *nrm = 0.0f;   // reset accumulator for this iteration (stream-ordered)
  if (t >= Nn * Nn) return;
  int j = t / Nn, a = t % Nn;
  const float* qr = Q + a * Nn;
  const float* xr = X + j * Nn;
  float mv = -1e30f;
  for (int b = 0; b < Nn; ++b) mv = fmaxf(mv, qr[b] * xr[b]);
  Mb[t] = mv;
}

// Xn = X*D + msg ; block-reduce sum of squares into *nrm
__global__ void k_update(const float* __restrict__ X, const float* __restrict__ Dm,
                         const float* __restrict__ msg, float* __restrict__ Xn,
                         float* __restrict__ nrm) {
  __shared__ float red[256];
  int t = blockIdx.x * blockDim.x + threadIdx.x;
  float v = 0.0f;
  if (t < Nn * Nn) {
    v = X[t] * Dm[t] + msg[t];
    Xn[t] = v;
  }
  red[threadIdx.x] = v * v;
  __syncthreads();
  for (int s = blockDim.x >> 1; s > 0; s >>= 1) {
    if ((int)threadIdx.x < s) red[threadIdx.x] += red[threadIdx.x + s];
    __syncthreads();
  }
  if (threadIdx.x == 0) atomicAdd(nrm, red[0]);
}

__global__ void k_normalize(const float* __restrict__ Xn, const float* __restrict__ nrm,
                            float* __restrict__ X) {
  int t = blockIdx.x * blockDim.x + threadIdx.x;
  if (t >= Nn * Nn) return;
  X[t] = Xn[t] * rsqrtf(*nrm);
}

// rec (BCE-with-logits over upper triangle) + kl ; out[0] = rec + kl
__global__ void k_loss(const float* __restrict__ vec, const float* __restrict__ Agt,
                       const float* __restrict__ mu, const float* __restrict__ lv,
                       float* __restrict__ out) {
  __shared__ float red[256];
  int tid = threadIdx.x;
  float acc = 0.0f;
  for (int t = tid; t < Nn * Nn; t += 256) {
    int i = t / Nn, j = t % Nn;
    if (i <= j) {
      float v  = vec[triidx(i, j)];
      float tt = Agt[t];
      acc += (fmaxf(v, 0.0f) - v * tt + log1pf(expf(-fabsf(v)))) / (float)TRI;
    }
  }
  for (int t = tid; t < ZD; t += 256) {
    float m = mu[t], l = lv[t];
    acc += (-0.5f / (float)ZD) * (1.0f + l - m * m - expf(l));
  }
  red[tid] = acc;
  __syncthreads();
  for (int s = 128; s > 0; s >>= 1) {
    if (tid < s) red[tid] += red[tid + s];
    __syncthreads();
  }
  if (tid == 0) out[0] = red[0];
}

__global__ void k_writeX(const float* __restrict__ X, float* __restrict__ out) {
  int t = blockIdx.x * blockDim.x + threadIdx.x;
  if (t >= Nn * Nn) return;
  out[t] = X[t];
}

// ---------------------------------------------------------------------------
extern "C" void kernel_launch(void* const* d_in, const int* in_sizes, int n_in,
                              void* d_out, int out_size, void* d_ws, size_t ws_size,
                              hipStream_t stream) {
  const float* x    = (const float*)d_in[0];
  const int*   ei   = (const int*)d_in[1];
  const float* Agt  = (const float*)d_in[3];
  const float* c1w  = (const float*)d_in[4];
  const float* c1b  = (const float*)d_in[5];
  const float* bn1g = (const float*)d_in[6];
  const float* bn1b = (const float*)d_in[7];
  const float* c2w  = (const float*)d_in[8];
  const float* c2b  = (const float*)d_in[9];
  const float* bn2g = (const float*)d_in[10];
  const float* bn2b = (const float*)d_in[11];
  const float* muw  = (const float*)d_in[12];
  const float* mub  = (const float*)d_in[13];
  const float* lvw  = (const float*)d_in[14];
  const float* lvb  = (const float*)d_in[15];
  const float* d1w  = (const float*)d_in[16];
  const float* d1b  = (const float*)d_in[17];
  const float* d2w  = (const float*)d_in[18];
  const float* d2b  = (const float*)d_in[19];
  const float* eps  = (const float*)d_in[20];
  float* out = (float*)d_out;

  // workspace carve-up (floats)
  float* ws    = (float*)d_ws;
  float* An    = ws;              // 9216
  float* H1raw = An + 9216;       // 24576
  float* Hagg  = H1raw + 24576;   // 24576
  float* H1    = Hagg + 24576;    // 24576
  float* H2raw = H1 + 24576;      // 24576
  float* H2    = H2raw + 24576;   // 24576
  float* g     = H2 + 24576;      // 256
  float* mu    = g + 256;         // 64
  float* lv    = mu + 64;         // 64
  float* z     = lv + 64;         // 64
  float* hpad  = z + 64;          // 16*256 = 4096
  float* vecP  = hpad + 4096;     // 16*4656 = 74496
  float* vec   = vecP + 74496;    // 4656
  float* Bm    = vec + 4656;      // 9216
  float* P     = Bm + 9216;       // 9216
  float* Q     = P + 9216;        // 9216
  float* Dm    = Q + 9216;        // 9216
  float* degA  = Dm + 9216;       // 96
  float* degB  = degA + 96;       // 96
  float* deg   = degB + 96;       // 96
  float* dis   = deg + 96;        // 96
  float* X     = dis + 96;        // 9216
  float* Xn    = X + 9216;        // 9216
  float* Mb    = Xn + 9216;       // 9216
  float* msg   = Mb + 9216;       // 9216
  float* nrm   = msg + 9216;      // 1

  // ---- encoder ----
  k_deg<<<1, 128, 0, stream>>>(ei, deg, dis);
  k_An_init<<<36, 256, 0, stream>>>(dis, An);
  k_An_edges<<<8, 256, 0, stream>>>(ei, dis, An);

  // H1raw = x @ W1  (96x64 * 64x256)
  k_wmma_gemm_f32<<<6 * 16, 32, 0, stream>>>(x, c1w, H1raw, HID, INF_, INF_, HID, HID);
  // Hagg = An @ H1raw  (96x96 * 96x256)
  k_wmma_gemm_f32<<<6 * 16, 32, 0, stream>>>(An, H1raw, Hagg, HID, Nn, Nn, HID, HID);
  k_bias_bn_relu<<<1, 256, 0, stream>>>(Hagg, c1b, bn1g, bn1b, H1);

  // H2raw = H1 @ W2  (96x256 * 256x256)
  k_wmma_gemm_f32<<<6 * 16, 32, 0, stream>>>(H1, c2w, H2raw, HID, HID, HID, HID, HID);
  k_wmma_gemm_f32<<<6 * 16, 32, 0, stream>>>(An, H2raw, Hagg, HID, Nn, Nn, HID, HID);
  k_bias_bn_relu<<<1, 256, 0, stream>>>(Hagg, c2b, bn2g, bn2b, H2);

  // ---- latent + decoder ----
  k_pool<<<1, 256, 0, stream>>>(H2, g);
  k_head<<<1, 64, 0, stream>>>(g, muw, mub, lvw, lvb, eps, mu, lv, z);
  k_dec1<<<16, 256, 0, stream>>>(z, d1w, d1b, hpad);
  // vecP = hpad @ dec2_w  (16x256 * 256x4656), only row 0 meaningful
  k_wmma_gemm_f32<<<1 * 291, 32, 0, stream>>>(hpad, d2w, vecP, TRI, HID, HID, TRI, TRI);
  k_buildB<<<36, 256, 0, stream>>>(vecP, d2b, vec, Bm);

  // ---- MPM setup ----
  k_degAB<<<1, 96, 0, stream>>>(Agt, Bm, degA, degB);
  k_pqdx<<<36, 256, 0, stream>>>(Agt, Bm, degA, degB, P, Q, Dm, X);

  // ---- MPM loop (rank-1 factorized: never materialize N^4 tensor) ----
  for (int it = 0; it < MPM_ITERS; ++it) {
    k_maxM<<<36, 256, 0, stream>>>(Q, X, Mb, nrm);
    // msg = P @ M  (96x96 * 96x96)
    k_wmma_gemm_f32<<<36, 32, 0, stream>>>(P, Mb, msg, Nn, Nn, Nn, Nn, Nn);
    k_update<<<36, 256, 0, stream>>>(X, Dm, msg, Xn, nrm);
    k_normalize<<<36, 256, 0, stream>>>(Xn, nrm, X);
  }

  // ---- outputs ----
  k_loss<<<1, 256, 0, stream>>>(vec, Agt, mu, lv, out);
  k_writeX<<<36, 256, 0, stream>>>(X, out + 1);
}